// BondWeight_41738492182540
// MI455X (gfx1250) — compile-verified
//
#include <hip/hip_runtime.h>

typedef unsigned int u32;
typedef unsigned long long u64;
typedef u32 v4u __attribute__((ext_vector_type(4)));
typedef int  v4i __attribute__((ext_vector_type(4)));
typedef int  v8i __attribute__((ext_vector_type(8)));

// Problem shape (fixed by the reference): B=1024, E=512, T=8, N=256.
#define NE 512
#define NN 256

// ---------------- TDM zero-fill ----------------
// Each workgroup: zero 32KB of LDS, then one TENSOR_STORE_FROM_LDS descriptor
// with iterate_enable=1 streams that tile 32x (lds_addr_increment=0,
// global_addr_increment = tile elements) => 1 MiB per workgroup.
// 256 workgroups => 256 MiB, spread across many WGPs/TDM engines so the
// aggregate can reach HBM store bandwidth (23.3 TB/s => ~11.5 us floor).
#define TDM_LDS_BYTES 32768
#define TDM_ITERS     32
#define TDM_CHUNK     ((u64)TDM_LDS_BYTES * (u64)TDM_ITERS)   // 1 MiB

__global__ void bw_fill_zero_tdm(float* __restrict__ out) {
  __shared__ unsigned long long zbuf[TDM_LDS_BYTES / 8];
  for (int i = threadIdx.x; i < TDM_LDS_BYTES / 8; i += blockDim.x)
    zbuf[i] = 0ull;
  __syncthreads();

  if (threadIdx.x == 0) {
    const u32 lds_addr = (u32)(size_t)(&zbuf[0]);
    const u64 gaddr = (((u64)(size_t)out) + (u64)blockIdx.x * TDM_CHUNK)
                      & ((1ULL << 57) - 1);

    const u32 TD0   = 0x7FFFFFFFu;          // tensor_dim0: huge -> no OOB clip
    const u32 TD1   = 1u;                   // tensor_dim1
    const u32 TILE0 = TDM_LDS_BYTES / 8;    // 4096 elements of 8B
    const u64 S0    = (u64)TILE0;           // tensor_dim0_stride

    // D# group 0: count=1 | lds_addr | global_addr | type=2
    v4u g0;
    g0.x = 0x1u;
    g0.y = lds_addr;
    g0.z = (u32)gaddr;
    g0.w = ((u32)(gaddr >> 32) & 0x01FFFFFFu) | (2u << 30);

    // D# group 1
    v8i g1;
    g1[0] = (int)((3u << 16) | (1u << 19));                 // data_size=8B, iterate_enable=1
    g1[1] = (int)((TD0 & 0xFFFFu) << 16);                   // tensor_dim0 lo16
    g1[2] = (int)(((TD0 >> 16) & 0xFFFFu) | ((TD1 & 0xFFFFu) << 16));
    g1[3] = (int)(((TD1 >> 16) & 0xFFFFu) | ((TILE0 & 0xFFFFu) << 16)); // tile_dim0
    g1[4] = (int)1u;                                        // tile_dim1=1, tile_dim2=0
    g1[5] = (int)(u32)(S0 & 0xFFFFFFFFu);                   // dim0 stride lo
    g1[6] = (int)(u32)((S0 >> 32) & 0xFFFFu);               // dim0 stride hi, dim1 stride=0
    g1[7] = 0;

    // D# group 2 (iterate mode): tensor_dim2=1, lds_inc=0, global_inc=TILE0 elems,
    // iterate_count = TDM_ITERS-1 (=> 32 iterations)
    v4i g2;
    g2[0] = 1;
    g2[1] = 0;
    g2[2] = (int)TILE0;
    g2[3] = (int)((u32)(TDM_ITERS - 1) << 16);

    v4i g3 = {0, 0, 0, 0};

    // Extra 8-dword group required by the 6-arg (clang-23 / amdgpu-toolchain)
    // form of the builtin; semantics not used here (zero-filled).
    v8i g4 = {0, 0, 0, 0, 0, 0, 0, 0};

    __builtin_amdgcn_tensor_store_from_lds(g0, g1, g2, g3, g4, 0);
    __builtin_amdgcn_s_wait_tensorcnt(0);
  }
}

// ---------------- deterministic scatter: tag pass ----------------
// Priority tag per write: pass0 (src,dst) -> e+1 ; pass1 (dst,src) -> NE+e+1.
// atomicMax is order-independent => deterministic winner = sequential last-wins,
// reproducing the reference's two scatter-set passes exactly, race-free.
__global__ void bw_tag(const int* __restrict__ src, const int* __restrict__ dst,
                       u32* __restrict__ out, int nedges) {
  int idx = blockIdx.x * blockDim.x + threadIdx.x;
  if (idx >= nedges) return;
  int b = idx / NE;
  int e = idx - b * NE;
  int s = src[idx] + 1;
  int d = dst[idx] + 1;
  u32* o = out + (size_t)b * NN * NN;
  atomicMax(&o[s * NN + d], (u32)(e + 1));
  atomicMax(&o[d * NN + s], (u32)(NE + e + 1));
}

// ---------------- deterministic scatter: resolve pass ----------------
// Exactly one winner per tagged cell writes the float bits; tags (1..1024)
// are denormal bit patterns disjoint from any weight value, so concurrent
// reads stay deterministic.
__global__ void bw_resolve(const float* __restrict__ w, const int* __restrict__ src,
                           const int* __restrict__ dst, const int* __restrict__ typ,
                           u32* __restrict__ out, int nedges) {
  int idx = blockIdx.x * blockDim.x + threadIdx.x;
  if (idx >= nedges) return;
  int b = idx / NE;
  int e = idx - b * NE;
  int s = src[idx] + 1;
  int d = dst[idx] + 1;
  u32 wbits = __float_as_uint(w[typ[idx]]);
  u32* o = out + (size_t)b * NN * NN;
  if (o[s * NN + d] == (u32)(e + 1))      o[s * NN + d] = wbits;
  if (o[d * NN + s] == (u32)(NE + e + 1)) o[d * NN + s] = wbits;
}

extern "C" void kernel_launch(void* const* d_in, const int* in_sizes, int n_in,
                              void* d_out, int out_size, void* d_ws, size_t ws_size,
                              hipStream_t stream) {
  const float* weights = (const float*)d_in[0];
  const int*   bsrc    = (const int*)d_in[1];
  const int*   bdst    = (const int*)d_in[2];
  const int*   btype   = (const int*)d_in[3];
  float* out = (float*)d_out;

  // 1) zero-fill 256 MiB via TDM: 1 MiB per workgroup -> 256 workgroups
  u64 total_bytes = (u64)out_size * 4ull;
  int nchunks = (int)(total_bytes / TDM_CHUNK);   // 268435456 / 1048576 = 256
  bw_fill_zero_tdm<<<dim3(nchunks), dim3(256), 0, stream>>>(out);

  // 2) + 3) deterministic symmetric scatter-set
  int nedges = in_sizes[1];                       // B*E = 524288
  int blocks = (nedges + 255) / 256;
  bw_tag<<<dim3(blocks), dim3(256), 0, stream>>>(bsrc, bdst, (u32*)out, nedges);
  bw_resolve<<<dim3(blocks), dim3(256), 0, stream>>>(weights, bsrc, bdst, btype,
                                                     (u32*)out, nedges);
}